// RHF_39977555591576
// MI455X (gfx1250) — compile-verified
//
#include <hip/hip_runtime.h>
#include <math.h>

typedef __attribute__((ext_vector_type(2))) float v2f;
typedef __attribute__((ext_vector_type(8))) float v8f;

#define PI_F 3.14159265358979323846f

__device__ __forceinline__ float boys0f(float x) {
    if (x < 1e-9f) return 1.0f - x * (1.0f / 3.0f);
    float sx = sqrtf(x);
    return 0.5f * sqrtf(PI_F / x) * erff(sx);
}

__device__ __forceinline__ float alpha_of(int i) {
    int m = i & 3;
    float a = 0.5f;
    a = (m == 1) ? 0.4f : a;
    a = (m == 2) ? 0.3f : a;
    a = (m == 3) ? 0.2f : a;
    return a;
}

// 8x8 fp32 matmul Dst = X @ Y (or X @ Y^T) using V_WMMA_F32_16X16X4_F32,
// padded to 16x16, K=8 done as two K=4 WMMA steps. Must be called by all
// 32 lanes in uniform control flow (EXEC all ones).
__device__ __forceinline__ void wmma_matmul8(float* __restrict__ Dst,
                                             const float* __restrict__ X,
                                             const float* __restrict__ Y,
                                             int lane, bool transY) {
    const int row   = lane & 15;          // A-row for this lane (16x4 tile)
    const int col   = lane & 15;          // B-col for this lane (4x16 tile)
    const int khalf = (lane >> 4) << 1;   // lanes 16-31 hold K offset +2

    v8f acc = {};
#pragma unroll
    for (int k0 = 0; k0 < 8; k0 += 4) {
        const int ka = k0 + khalf;        // ka, ka+1 in [0,8)
        // A operand: 16x4 fp32 -> 2 VGPRs/lane. Rows 8..15 are zero padding.
        float ax = X[(row & 7) * 8 + ka];
        float ay = X[(row & 7) * 8 + ka + 1];
        ax = (row < 8) ? ax : 0.0f;
        ay = (row < 8) ? ay : 0.0f;
        // B operand: 4x16 fp32 -> 2 VGPRs/lane. Cols 8..15 are zero padding.
        float bx, by;
        if (!transY) {
            bx = Y[ka * 8 + (col & 7)];
            by = Y[(ka + 1) * 8 + (col & 7)];
        } else {
            bx = Y[(col & 7) * 8 + ka];
            by = Y[(col & 7) * 8 + ka + 1];
        }
        bx = (col < 8) ? bx : 0.0f;
        by = (col < 8) ? by : 0.0f;

        v2f a; a.x = ax; a.y = ay;
        v2f b; b.x = bx; b.y = by;
        acc = __builtin_amdgcn_wmma_f32_16x16x4_f32(
            /*neg_a=*/false, a, /*neg_b=*/false, b,
            /*c_mod=*/(short)0, acc, /*reuse_a=*/false, /*reuse_b=*/false);
    }
    // D layout: VGPR v holds row v (lanes 0-15) / row v+8 (lanes 16-31).
    // Only rows 0..7, cols 0..7 are meaningful.
    if (lane < 8) {
#pragma unroll
        for (int v = 0; v < 8; ++v) Dst[v * 8 + lane] = acc[v];
    }
}

// Cyclic Jacobi eigensolver for an 8x8 symmetric matrix in LDS.
// On exit: ev[] ascending eigenvalues, Vm columns = matching eigenvectors.
// Run by a single lane.
__device__ void jacobi8(float* M, float* Vm, float* ev) {
    for (int i = 0; i < 8; ++i)
        for (int j = 0; j < 8; ++j)
            Vm[i * 8 + j] = (i == j) ? 1.0f : 0.0f;

    for (int sweep = 0; sweep < 14; ++sweep) {
        for (int i = 0; i < 7; ++i) {
            for (int j = i + 1; j < 8; ++j) {
                float apq = M[i * 8 + j];
                if (fabsf(apq) <= 1e-13f) continue;
                float app = M[i * 8 + i];
                float aqq = M[j * 8 + j];
                float theta = 0.5f * (aqq - app) / apq;
                float t = 1.0f / (fabsf(theta) + sqrtf(1.0f + theta * theta));
                t = (theta >= 0.0f) ? t : -t;
                float c = 1.0f / sqrtf(1.0f + t * t);
                float s = t * c;
                for (int k = 0; k < 8; ++k) {
                    float aik = M[i * 8 + k], ajk = M[j * 8 + k];
                    M[i * 8 + k] = c * aik - s * ajk;
                    M[j * 8 + k] = s * aik + c * ajk;
                }
                for (int k = 0; k < 8; ++k) {
                    float aki = M[k * 8 + i], akj = M[k * 8 + j];
                    M[k * 8 + i] = c * aki - s * akj;
                    M[k * 8 + j] = s * aki + c * akj;
                }
                for (int k = 0; k < 8; ++k) {
                    float vki = Vm[k * 8 + i], vkj = Vm[k * 8 + j];
                    Vm[k * 8 + i] = c * vki - s * vkj;
                    Vm[k * 8 + j] = s * vki + c * vkj;
                }
            }
        }
    }
    for (int i = 0; i < 8; ++i) ev[i] = M[i * 8 + i];
    // sort ascending, permute eigenvector columns
    for (int i = 0; i < 7; ++i) {
        int m = i;
        for (int j = i + 1; j < 8; ++j)
            if (ev[j] < ev[m]) m = j;
        if (m != i) {
            float tv = ev[i]; ev[i] = ev[m]; ev[m] = tv;
            for (int k = 0; k < 8; ++k) {
                float tc = Vm[k * 8 + i];
                Vm[k * 8 + i] = Vm[k * 8 + m];
                Vm[k * 8 + m] = tc;
            }
        }
    }
}

__global__ __launch_bounds__(32)
void rhf_wave_kernel(const float* __restrict__ geom, float* __restrict__ out) {
    __shared__ float sG[4096];    // G[p][q][r][s]  -> J contraction
    __shared__ float sGK[4096];   // G[p][r][q][s]  -> K contraction
    __shared__ float sH[64], sS[64], sA[64], sF[64];
    __shared__ float sT1[64], sFp[64], sC2[64], sC[64], sD[64], sB1[64];
    __shared__ float sNN[64], sPab[64], sKab[64], sP[64 * 3];
    __shared__ float sEv[8], sRed[32];

    const int lane = threadIdx.x;

    // ---- geometry (broadcast scalar loads) ----
    const float gx0 = geom[0], gy0 = geom[1], gz0 = geom[2];
    const float gx1 = geom[3], gy1 = geom[4], gz1 = geom[5];
    const float ddx = gx0 - gx1, ddy = gy0 - gy1, ddz = gz0 - gz1;
    const float Enuc = 1.0f / sqrtf(ddx * ddx + ddy * ddy + ddz * ddz); // Z0*Z1 = 1

    // ---- one- and two-center pair quantities: S, T, V, H, and G precursors ----
#pragma unroll
    for (int rep = 0; rep < 2; ++rep) {
        int t = lane + rep * 32;          // pair index p*8 + q
        int p = t >> 3, q = t & 7;
        float ap = alpha_of(p), aq = alpha_of(q);
        float pab = ap + aq;
        float cpx = (p < 4) ? gx0 : gx1, cpy = (p < 4) ? gy0 : gy1, cpz = (p < 4) ? gz0 : gz1;
        float cqx = (q < 4) ? gx0 : gx1, cqy = (q < 4) ? gy0 : gy1, cqz = (q < 4) ? gz0 : gz1;
        float abx = cpx - cqx, aby = cpy - cqy, abz = cpz - cqz;
        float AB2 = abx * abx + aby * aby + abz * abz;
        float Np = powf(2.0f * ap / PI_F, 0.75f);
        float Nq = powf(2.0f * aq / PI_F, 0.75f);
        float NN = Np * Nq;
        float mu = ap * aq / pab;
        float Kab = expf(-mu * AB2);
        float Spq = NN * powf(PI_F / pab, 1.5f) * Kab;
        float Tpq = Spq * mu * (3.0f - 2.0f * mu * AB2);
        float Px = (ap * cpx + aq * cqx) / pab;
        float Py = (ap * cpy + aq * cqy) / pab;
        float Pz = (ap * cpz + aq * cqz) / pab;
        float d0x = Px - gx0, d0y = Py - gy0, d0z = Pz - gz0;
        float d1x = Px - gx1, d1y = Py - gy1, d1z = Pz - gz1;
        float pc0 = d0x * d0x + d0y * d0y + d0z * d0z;
        float pc1 = d1x * d1x + d1y * d1y + d1z * d1z;
        float Vpq = -NN * (2.0f * PI_F / pab) * Kab *
                    (boys0f(pab * pc0) + boys0f(pab * pc1));
        sS[t] = Spq;
        sH[t] = Tpq + Vpq;
        sNN[t] = NN; sPab[t] = pab; sKab[t] = Kab;
        sP[t * 3 + 0] = Px; sP[t * 3 + 1] = Py; sP[t * 3 + 2] = Pz;
    }
    __syncthreads();

    // ---- two-electron tensor G (and its exchange permutation) into LDS ----
    const float TwoPi25 = 34.986836655249725f;  // 2 * pi^2.5
    for (int i = 0; i < 128; ++i) {
        int idx = (i << 5) + lane;        // [0, 4096)
        int pqi = idx >> 6;               // p*8+q
        int rsi = idx & 63;               // r*8+s
        float pp = sPab[pqi], qq = sPab[rsi];
        float dxp = sP[pqi * 3 + 0] - sP[rsi * 3 + 0];
        float dyp = sP[pqi * 3 + 1] - sP[rsi * 3 + 1];
        float dzp = sP[pqi * 3 + 2] - sP[rsi * 3 + 2];
        float PQ2 = dxp * dxp + dyp * dyp + dzp * dzp;
        float psum = pp + qq;
        float val = sNN[pqi] * sNN[rsi] * TwoPi25 /
                    (pp * qq * sqrtf(psum)) * sKab[pqi] * sKab[rsi] *
                    boys0f(pp * qq / psum * PQ2);
        sG[idx] = val;
        int p = idx >> 9, q = (idx >> 6) & 7, r = (idx >> 3) & 7, s = idx & 7;
        sGK[((p * 8 + r) * 8 + q) * 8 + s] = val;   // GK[p][r'][q'][s] = G[p][q'][r'][s]
    }
    __syncthreads();

    // ---- eigh(S) -> orthogonalizer A = svec * sval^-1/2 @ svec^T ----
    if (lane == 0) {
        for (int t = 0; t < 64; ++t) sT1[t] = sS[t];
        jacobi8(sT1, sC2, sEv);
    }
    __syncthreads();
#pragma unroll
    for (int rep = 0; rep < 2; ++rep) {
        int t = lane + rep * 32;
        sB1[t] = sC2[t] * rsqrtf(sEv[t & 7]);   // column scaling by sval^-0.5
        sD[t] = 0.0f;                            // initial density
    }
    __syncthreads();
    wmma_matmul8(sA, sB1, sC2, lane, /*transY=*/true);
    __syncthreads();

    // ---- SCF iterations ----
    for (int it = 0; it < 20; ++it) {
        // F = H + 2J - K  (J/K: 64x64 mat-vec against flattened D)
#pragma unroll
        for (int rep = 0; rep < 2; ++rep) {
            int t = lane + rep * 32;
            const float* gp  = &sG[t * 64];
            const float* gkp = &sGK[t * 64];
            float Jv = 0.0f, Kv = 0.0f;
            for (int u = 0; u < 64; ++u) {
                float d = sD[u];
                Jv = fmaf(gp[u], d, Jv);
                Kv = fmaf(gkp[u], d, Kv);
            }
            sF[t] = sH[t] + 2.0f * Jv - Kv;
        }
        __syncthreads();

        // Fp = A @ F @ A   (fp32 WMMA)
        wmma_matmul8(sT1, sA, sF, lane, false);
        __syncthreads();
        wmma_matmul8(sFp, sT1, sA, lane, false);
        __syncthreads();

        // eigh(Fp) -> eps, C2
        if (lane == 0) jacobi8(sFp, sC2, sEv);
        __syncthreads();

        // C = A @ C2       (fp32 WMMA)
        wmma_matmul8(sC, sA, sC2, lane, false);
        __syncthreads();

        // D = Cocc @ Cocc^T  (NDOCC = 1 -> rank-1, lowest-eps column)
#pragma unroll
        for (int rep = 0; rep < 2; ++rep) {
            int t = lane + rep * 32;
            sD[t] = sC[(t >> 3) * 8] * sC[(t & 7) * 8];
        }
        __syncthreads();
    }

    // ---- E_scf = sum((F + H) .* D) + Enuc ----
    float partial = 0.0f;
#pragma unroll
    for (int rep = 0; rep < 2; ++rep) {
        int t = lane + rep * 32;
        partial += (sF[t] + sH[t]) * sD[t];
    }
    sRed[lane] = partial;
    __syncthreads();
    if (lane == 0) {
        float e = Enuc;
        for (int k = 0; k < 32; ++k) e += sRed[k];
        out[0] = e;
        for (int k = 0; k < 8; ++k) out[1 + k] = sEv[k];
    }
#pragma unroll
    for (int rep = 0; rep < 2; ++rep) {
        int t = lane + rep * 32;
        out[9 + t] = sC[t];
    }
}

extern "C" void kernel_launch(void* const* d_in, const int* in_sizes, int n_in,
                              void* d_out, int out_size, void* d_ws, size_t ws_size,
                              hipStream_t stream) {
    (void)in_sizes; (void)n_in; (void)d_ws; (void)ws_size; (void)out_size;
    const float* geom = (const float*)d_in[0];   // 6 floats; d_in[1] (row_idx) unused
    float* out = (float*)d_out;                  // 73 floats: E, eps[8], C[64]
    rhf_wave_kernel<<<dim3(1), dim3(32), 0, stream>>>(geom, out);
}